// DifferentiableTopologyRegularizer_49881750176381
// MI455X (gfx1250) — compile-verified
//
#include <hip/hip_runtime.h>

// Problem constants (match reference setup_inputs)
#define B_DIM   128
#define T_DIM   2048
#define D_DIM   512
#define SUB     128          // SUBSAMPLE
#define RSTRIDE 16           // T_DIM / SUB
#define ROWELT  (RSTRIDE * D_DIM)   // 8192 elements between subsampled rows
#define KCHUNK  64
#define NCHUNK  (D_DIM / KCHUNK)    // 8
#define LSTRIDE (KCHUNK + 4)        // 68 floats: bank = (4*row + k) % 64 -> conflict-free frags
#define NTRIP   32
#define NTHREADS 1024

typedef __attribute__((ext_vector_type(2))) float v2f;
typedef __attribute__((ext_vector_type(8))) float v8f;
typedef __attribute__((ext_vector_type(4))) unsigned int v4u;
typedef __attribute__((ext_vector_type(8))) int v8i;
typedef __attribute__((ext_vector_type(4))) int v4i;

// addrspace(3) (LDS) qualified types: keep reads on the ds_* path
typedef __attribute__((address_space(3))) float lds_f;
typedef __attribute__((address_space(3))) v2f   lds_v2f;

// Issue one TDM 2D tile load: 128 rows x 64 floats, row stride 8192 floats,
// with HW padding of 4 DWORDs after every 64 DWORDs -> LDS row stride 68 floats.
__device__ __forceinline__ void tdm_load_chunk(const float* gsrc, unsigned lds_byte_off)
{
    unsigned long long ga = (unsigned long long)gsrc;
    v4u g0;
    g0.x = 1u;                                   // count=1, is_restore=0, gather off
    g0.y = lds_byte_off;                         // LDS dest byte address
    g0.z = (unsigned)(ga & 0xFFFFFFFFu);         // global_addr[31:0]  (bits 95:64)
    g0.w = (unsigned)((ga >> 32) & 0x01FFFFFFu)  // global_addr[56:32] (bits 120:96)
           | (2u << 30);                         // type=2 ("image")   (bits 127:126)

    v8i g1;
    g1[0] = (2 << 16)      // data_size = 2 -> 4 bytes
          | (1 << 20)      // pad_enable
          | (5 << 22)      // pad_interval = 5 -> every 64 DWORDs
          | (3 << 25);     // pad_amount   = 3 -> 4 DWORDs of pad
    g1[1] = (KCHUNK << 16);        // tensor_dim0[15:0] = 64 (bits 63:48)
    g1[2] = (SUB << 16);           // tensor_dim0[31:16]=0 | tensor_dim1[15:0]=128
    g1[3] = (KCHUNK << 16);        // tensor_dim1[31:16]=0 | tile_dim0=64
    g1[4] = SUB;                   // tile_dim1=128 | tile_dim2=0
    g1[5] = ROWELT;                // tensor_dim0_stride[31:0] = 8192 elements
    g1[6] = 0;                     // stride[47:32]=0 | tensor_dim1_stride lo = 0
    g1[7] = 0;

    v4i gz4 = {0, 0, 0, 0};            // groups 2/3: dims >2 unused
    v8i gz8 = {0, 0, 0, 0, 0, 0, 0, 0};

    // 6-arg toolchain variant: (g0, g1, g2, g3, g4, cpol)
    __builtin_amdgcn_tensor_load_to_lds(g0, g1, gz4, gz4, gz8, 0);
}

__global__ __launch_bounds__(NTHREADS) void topo_loss_kernel(
    const float* __restrict__ latent,       // [B, T, D] fp32
    const float* __restrict__ conn_thresh,  // [1] fp32
    const int*   __restrict__ trip_idx,     // [B, 32, 3] int32, values in [0,128)
    float* __restrict__ per_batch)          // [B] partial losses
{
    __shared__ float lds_x[2][SUB][LSTRIDE];   // double-buffered chunk: 2 x 34 KB
    __shared__ float sqn[SUB];
    __shared__ float red[NTHREADS];
    __shared__ float edge_d[3 * NTRIP];
    __shared__ float hole[NTRIP];

    const int b    = blockIdx.x;
    const int tid  = threadIdx.x;
    const int lane = tid & 31;
    const int wv   = tid >> 5;        // wave id 0..31
    const int lhi  = lane >> 4;       // 0/1: K-half of the fragment
    const int l16  = lane & 15;

    // tile ownership: 64 tiles (8x8), 2 per wave
    const int iblk = wv >> 2;              // row block 0..7
    const int j0   = (wv & 3) * 2;         // col blocks j0, j0+1
    const int j1   = j0 + 1;

    const float* xb = latent + (size_t)b * T_DIM * D_DIM;

    // LDS byte offsets of the two chunk buffers (for the TDM descriptor)
    lds_f* xl = (lds_f*)&lds_x[0][0][0];
    // launder the 32-bit LDS pointer: buffer is written only by the TDM engine;
    // keeping it addrspace(3) keeps all reads on the ds_load_* path.
    asm volatile("" : "+v"(xl) :: "memory");
    const unsigned lds_base  = (unsigned)(unsigned long long)xl;
    const unsigned buf_bytes = SUB * LSTRIDE * 4;

    v8f acc0 = {};
    v8f acc1 = {};
    float sqn_acc = 0.0f;

    // Triplet edge assignment: thread e < 96 owns edge e = 3*t + p
    int ea = 0, ec = 0;
    float edot = 0.0f;
    if (tid < 96) {
        int t = tid / 3;
        int p = tid - 3 * t;
        const int* ti = trip_idx + ((size_t)b * NTRIP + t) * 3;
        int i0 = ti[0], i1 = ti[1], i2 = ti[2];
        ea = (p == 2) ? i1 : i0;          // edges: (i0,i1), (i0,i2), (i1,i2)
        ec = (p == 0) ? i1 : i2;
    }

    // prologue: start chunk 0 DMA
    if (wv == 0) {
        tdm_load_chunk(xb, lds_base);
    }

    for (int c = 0; c < NCHUNK; ++c) {
        const int cb = c & 1;
        if (wv == 0) {
            if (c + 1 < NCHUNK) {
                tdm_load_chunk(xb + (c + 1) * KCHUNK,
                               lds_base + ((c + 1) & 1) * buf_bytes);
                __builtin_amdgcn_s_wait_tensorcnt((short)1);  // chunk c done, c+1 in flight
            } else {
                __builtin_amdgcn_s_wait_tensorcnt((short)0);  // last chunk done
            }
        }
        __syncthreads();

        const lds_f* buf = xl + cb * (SUB * LSTRIDE);

        // ---- Gram accumulation: 16 K-steps of 4 via fp32 WMMA ----
        {
            const lds_f* arow  = buf + (iblk * 16 + l16) * LSTRIDE;
            const lds_f* brow0 = buf + (j0 * 16 + l16) * LSTRIDE;
            const lds_f* brow1 = buf + (j1 * 16 + l16) * LSTRIDE;
            #pragma unroll 4
            for (int kk = 0; kk < KCHUNK / 4; ++kk) {
                const int kloc = kk * 4 + lhi * 2;  // lane<16 -> K{k0,k0+1}; lane>=16 -> K{k0+2,k0+3}
                v2f afrag  = *(const lds_v2f*)(arow + kloc);
                v2f bfrag0 = *(const lds_v2f*)(brow0 + kloc);
                v2f bfrag1 = *(const lds_v2f*)(brow1 + kloc);
                acc0 = __builtin_amdgcn_wmma_f32_16x16x4_f32(
                    false, afrag, false, bfrag0, (short)0, acc0, false, false);
                acc1 = __builtin_amdgcn_wmma_f32_16x16x4_f32(
                    false, afrag, false, bfrag1, (short)0, acc1, false, false);
            }
        }

        // ---- squared norms: thread t<128 owns row t ----
        if (tid < SUB) {
            const lds_f* row = buf + tid * LSTRIDE;
            float s = 0.0f;
            #pragma unroll 8
            for (int k = 0; k < KCHUNK; ++k) {
                float v = row[k];
                s = fmaf(v, v, s);
            }
            sqn_acc += s;
        }

        // ---- triplet edge dot products ----
        if (tid < 96) {
            const lds_f* ra = buf + ea * LSTRIDE;
            const lds_f* rc = buf + ec * LSTRIDE;
            float s = 0.0f;
            #pragma unroll 8
            for (int k = 0; k < KCHUNK; ++k)
                s = fmaf(ra[k], rc[k], s);
            edot += s;
        }
        __syncthreads();
    }

    if (tid < SUB) sqn[tid] = sqn_acc;
    __syncthreads();

    const float th = fabsf(conn_thresh[0]) + 0.1f;

    // ---- connectivity: per-lane sigmoid sums, diagonal excluded ----
    // C/D layout: lanes 0-15: VGPR v -> M = v; lanes 16-31: M = v + 8; N = lane%16.
    float conn = 0.0f;
    {
        const int n0 = j0 * 16 + l16;
        const int n1 = j1 * 16 + l16;
        const float sn0 = sqn[n0];
        const float sn1 = sqn[n1];
        #pragma unroll
        for (int v = 0; v < 8; ++v) {
            const int m = iblk * 16 + v + 8 * lhi;
            const float sm = sqn[m];
            float sq0 = fmaxf(sm + sn0 - 2.0f * acc0[v], 0.0f);
            float sq1 = fmaxf(sm + sn1 - 2.0f * acc1[v], 0.0f);
            float d0 = sqrtf(sq0);
            float d1 = sqrtf(sq1);
            if (m != n0) conn += 1.0f / (1.0f + expf(d0 - th));
            if (m != n1) conn += 1.0f / (1.0f + expf(d1 - th));
        }
    }
    red[tid] = conn;

    if (tid < 96) {
        float sq = fmaxf(sqn[ea] + sqn[ec] - 2.0f * edot, 0.0f);
        edge_d[tid] = sqrtf(sq);
    }
    __syncthreads();

    if (tid < NTRIP) {
        float e0 = edge_d[3 * tid + 0];
        float e1 = edge_d[3 * tid + 1];
        float e2 = edge_d[3 * tid + 2];
        float mu = (e0 + e1 + e2) * (1.0f / 3.0f);
        float va = ((e0 - mu) * (e0 - mu) + (e1 - mu) * (e1 - mu) +
                    (e2 - mu) * (e2 - mu)) * 0.5f;   // ddof = 1
        hole[tid] = expf(-va);
    }

    // deterministic fixed-order tree reduction of conn over 1024 threads
    __syncthreads();
    #pragma unroll
    for (int s = NTHREADS / 2; s >= 1; s >>= 1) {
        if (tid < s) red[tid] += red[tid + s];
        __syncthreads();
    }

    if (tid == 0) {
        const float denom = (float)(SUB * (SUB - 1)) + 1e-8f;
        float connectivity_loss = 1.0f - red[0] / denom;

        float hole_sum = 0.0f;
        for (int i = 0; i < NTRIP; ++i) hole_sum += hole[i];  // fixed order
        float hole_loss = hole_sum * (1.0f / NTRIP);

        per_batch[b] = connectivity_loss + 0.5f * hole_loss;
    }
}

__global__ void topo_reduce_kernel(const float* __restrict__ per_batch,
                                   float* __restrict__ out)
{
    if (blockIdx.x == 0 && threadIdx.x == 0) {
        float s = 0.0f;
        for (int i = 0; i < B_DIM; ++i) s += per_batch[i];    // fixed order: deterministic
        out[0] = s * (1.0f / B_DIM);
    }
}

extern "C" void kernel_launch(void* const* d_in, const int* in_sizes, int n_in,
                              void* d_out, int out_size, void* d_ws, size_t ws_size,
                              hipStream_t stream)
{
    (void)in_sizes; (void)n_in; (void)out_size; (void)ws_size;
    const float* latent = (const float*)d_in[0];
    const float* thresh = (const float*)d_in[1];
    const int*   trip   = (const int*)d_in[2];
    float* per_batch = (float*)d_ws;   // 128 floats of scratch

    topo_loss_kernel<<<B_DIM, NTHREADS, 0, stream>>>(latent, thresh, trip, per_batch);
    topo_reduce_kernel<<<1, 32, 0, stream>>>(per_batch, (float*)d_out);
}